// NeuralMemory_12567074308564
// MI455X (gfx1250) — compile-verified
//
#include <hip/hip_runtime.h>
#include <hip/hip_bf16.h>
#include <cstdint>
#include <cstddef>

// ---------------- problem constants ----------------
#define B_      2
#define S_      4096
#define DIM_    512
#define HEADS_  8
#define DH_     64
#define CHUNK_  64
#define NC_     64                 // S_/CHUNK_
#define BH_     (B_*HEADS_)        // 16
#define NCHUNKS_ (BH_*NC_)         // 1024
#define CURT_   (S_ - CHUNK_)      // 4032
#define EPS_    1e-6f
#define SCALE_  0.125f             // dh^-0.5

typedef float v2f __attribute__((ext_vector_type(2)));
typedef float v8f __attribute__((ext_vector_type(8)));

__device__ __forceinline__ float sigmoidf_(float x){ return 1.f/(1.f+__expf(-x)); }
__device__ __forceinline__ float siluf_(float x){ return x*sigmoidf_(x); }

// =====================================================================
// 64x64x64 matmul, 4 waves (128 threads), fp32 WMMA 16x16x4.
// D = scale * op(A) @ op(B), stride 64 everywhere (LDS or global D).
// A-fragment (16x4): lanes 0-15 row M, K={k0,k0+1}; lanes 16-31 K={k0+2,k0+3}
// B-fragment (4x16): lanes 0-15 col N, K rows likewise split by lane half.
// C/D: VGPR i -> M = m0 + i (+8 for upper lane half), N = lane&15.
// =====================================================================
template<bool TA, bool TB, bool SILU_A>
__device__ __forceinline__ void mm64(float* __restrict__ D,
                                     const float* __restrict__ A,
                                     const float* __restrict__ B,
                                     float scale)
{
    const int lane = threadIdx.x & 31;
    const int wave = threadIdx.x >> 5;
    const int m0   = wave * 16;
    const int l16  = lane & 15;
    const int kh   = (lane >> 4) << 1;   // 0 or 2
#pragma unroll
    for (int n0 = 0; n0 < 64; n0 += 16) {
        v8f acc = {};
#pragma unroll
        for (int k0 = 0; k0 < 64; k0 += 4) {
            const int ka = k0 + kh;
            v2f a, b;
            if (TA) { a.x = A[ka*64 + (m0+l16)];     a.y = A[(ka+1)*64 + (m0+l16)]; }
            else    { a.x = A[(m0+l16)*64 + ka];     a.y = A[(m0+l16)*64 + ka+1];   }
            if (SILU_A) { a.x = siluf_(a.x); a.y = siluf_(a.y); }
            if (TB) { b.x = B[(n0+l16)*64 + ka];     b.y = B[(n0+l16)*64 + ka+1];   }
            else    { b.x = B[ka*64 + (n0+l16)];     b.y = B[(ka+1)*64 + (n0+l16)]; }
            acc = __builtin_amdgcn_wmma_f32_16x16x4_f32(
                      false, a, false, b, (short)0, acc, false, false);
        }
        const int row0 = m0 + ((lane >> 4) ? 8 : 0);
        const int col  = n0 + l16;
#pragma unroll
        for (int i = 0; i < 8; ++i)
            D[(size_t)(row0 + i)*64 + col] = acc[i] * scale;
    }
}

// 64x64 tile MAC for K=512 GEMMs: acc[4] persists across k-blocks.
__device__ __forceinline__ void tile_mma(const float* __restrict__ A,
                                         const float* __restrict__ Bm,
                                         v8f* __restrict__ acc)
{
    const int lane = threadIdx.x & 31;
    const int wave = threadIdx.x >> 5;
    const int m0   = wave * 16;
    const int l16  = lane & 15;
    const int kh   = (lane >> 4) << 1;
#pragma unroll
    for (int t = 0; t < 4; ++t) {
#pragma unroll
        for (int k0 = 0; k0 < 64; k0 += 4) {
            const int ka = k0 + kh;
            v2f a, b;
            a.x = A[(m0+l16)*64 + ka];        a.y = A[(m0+l16)*64 + ka+1];
            b.x = Bm[ka*64 + t*16 + l16];     b.y = Bm[(ka+1)*64 + t*16 + l16];
            acc[t] = __builtin_amdgcn_wmma_f32_16x16x4_f32(
                         false, a, false, b, (short)0, acc[t], false, false);
        }
    }
}

// =====================================================================
// Kernel 1: dual rmsnorm -> s (store-normed) and r (retrieve-normed)
// one thread per row of [B,S,DIM]
// =====================================================================
__global__ void nm_norm_kernel(const float* __restrict__ seq,
                               const float* __restrict__ w_store,
                               const float* __restrict__ w_retr,
                               float* __restrict__ s, float* __restrict__ r)
{
    const int row = blockIdx.x * blockDim.x + threadIdx.x;     // 0..B*S-1
    if (row >= B_*S_) return;
    const float* x = seq + (size_t)row * DIM_;
    float ss = 0.f;
    for (int d = 0; d < DIM_; ++d) { float v = x[d]; ss += v*v; }
    const float inv = rsqrtf(ss * (1.f/DIM_) + EPS_);
    float* so = s + (size_t)row * DIM_;
    float* ro = r + (size_t)row * DIM_;
    for (int d = 0; d < DIM_; ++d) {
        const float v = x[d] * inv;
        so[d] = v * w_store[d];
        ro[d] = v * w_retr[d];
    }
}

// =====================================================================
// Kernel 2: per-(b,chunk) gates: lr[bh][ci], mom_gate, decay
// =====================================================================
__global__ void nm_gates_kernel(const float* __restrict__ s,
                                const float* __restrict__ w_step,
                                const float* __restrict__ w_mom,
                                const float* __restrict__ w_decay,
                                float* __restrict__ lr, float* __restrict__ mom,
                                float* __restrict__ dec)
{
    const int blk = blockIdx.x;              // 0..B*NC-1
    const int bb  = blk / NC_;
    const int ci  = blk % NC_;
    const float* sc = s + ((size_t)bb*S_ + (size_t)ci*CHUNK_) * DIM_;
    __shared__ float smean[DIM_];
    __shared__ float lracc[HEADS_];
    const int tid = threadIdx.x;
    for (int d = tid; d < DIM_; d += blockDim.x) {
        float acc = 0.f;
        for (int rr = 0; rr < CHUNK_; ++rr) acc += sc[(size_t)rr*DIM_ + d];
        smean[d] = acc * (1.f/CHUNK_);
    }
    if (tid < HEADS_) lracc[tid] = 0.f;
    __syncthreads();
    for (int p = tid; p < HEADS_*CHUNK_; p += blockDim.x) {
        const int h = p >> 6, rr = p & 63;
        float dot = 0.f;
        for (int d = 0; d < DIM_; ++d) dot += sc[(size_t)rr*DIM_ + d] * w_step[d*HEADS_ + h];
        atomicAdd(&lracc[h], __expf(sigmoidf_(dot) * -15.f));
    }
    __syncthreads();
    if (tid < HEADS_) {
        const int h = tid;
        const int o = (bb*HEADS_ + h)*NC_ + ci;
        lr[o] = lracc[h] * (1.f/CHUNK_);
        float dm = 0.f, dd = 0.f;
        for (int d = 0; d < DIM_; ++d) {
            dm += smean[d] * w_mom[d*HEADS_ + h];
            dd += smean[d] * w_decay[d*HEADS_ + h];
        }
        mom[o] = sigmoidf_(dm);
        dec[o] = sigmoidf_(dd);
    }
}

// =====================================================================
// Kernel 3: keys/values = s @ w_kv  (8192 x 1024 x 512), chunkified store
// =====================================================================
__global__ void nm_gemm_kv(const float* __restrict__ s, const float* __restrict__ w_kv,
                           float* __restrict__ keys, float* __restrict__ values)
{
    extern __shared__ float sm[];
    float* At = sm;            // 64x64
    float* Wt = sm + 4096;     // 64x64
    const int mt = blockIdx.x, nt = blockIdx.y, tid = threadIdx.x;
    v8f acc[4] = {};
    for (int kb = 0; kb < DIM_/64; ++kb) {
        for (int e = tid; e < 4096; e += 128) {
            const int rr = e >> 6, cc = e & 63;
            At[e] = s[(size_t)(mt*64 + rr)*DIM_ + kb*64 + cc];
            Wt[e] = w_kv[(size_t)(kb*64 + rr)*(2*DIM_) + nt*64 + cc];
        }
        __syncthreads();
        tile_mma(At, Wt, acc);
        __syncthreads();
    }
    const int lane = tid & 31, wave = tid >> 5;
    const int row0 = wave*16 + ((lane >> 4) ? 8 : 0);
    const int l16  = lane & 15;
#pragma unroll
    for (int t = 0; t < 4; ++t)
#pragma unroll
        for (int i = 0; i < 8; ++i) {
            const int grow = mt*64 + row0 + i;          // row in [0, B*S)
            const int gcol = nt*64 + t*16 + l16;        // col in [0, 1024)
            const int bb = grow >> 12, tt = grow & 4095;
            const int ci = tt >> 6, pos = tt & 63;
            const int isv = gcol >> 9;
            const int h  = (gcol >> 6) & 7;
            const int dc = gcol & 63;
            const size_t out = (((size_t)(bb*HEADS_ + h)*NC_ + ci)*CHUNK_ + pos)*DH_ + dc;
            (isv ? values : keys)[out] = acc[t][i];
        }
}

// =====================================================================
// Kernel 4: q = rc @ w_q  with rc = shifted/padded retrieve-normed seq
// =====================================================================
__global__ void nm_gemm_q(const float* __restrict__ r, const float* __restrict__ w_q,
                          float* __restrict__ qch)
{
    extern __shared__ float sm[];
    float* At = sm; float* Wt = sm + 4096;
    const int mt = blockIdx.x, nt = blockIdx.y, tid = threadIdx.x;
    v8f acc[4] = {};
    for (int kb = 0; kb < DIM_/64; ++kb) {
        for (int e = tid; e < 4096; e += 128) {
            const int rr = e >> 6, cc = e & 63;
            const int grow = mt*64 + rr;
            const int bb = grow >> 12, j = grow & 4095;
            At[e] = (j < CURT_) ? r[((size_t)bb*S_ + j + CHUNK_)*DIM_ + kb*64 + cc] : 0.f;
            Wt[e] = w_q[(size_t)(kb*64 + rr)*DIM_ + nt*64 + cc];
        }
        __syncthreads();
        tile_mma(At, Wt, acc);
        __syncthreads();
    }
    const int lane = tid & 31, wave = tid >> 5;
    const int row0 = wave*16 + ((lane >> 4) ? 8 : 0);
    const int l16  = lane & 15;
#pragma unroll
    for (int t = 0; t < 4; ++t)
#pragma unroll
        for (int i = 0; i < 8; ++i) {
            const int grow = mt*64 + row0 + i;
            const int gcol = nt*64 + t*16 + l16;        // [0,512)
            const int bb = grow >> 12, j = grow & 4095;
            const int ci = j >> 6, pos = j & 63;
            const int h = gcol >> 6, dc = gcol & 63;
            qch[(((size_t)(bb*HEADS_ + h)*NC_ + ci)*CHUNK_ + pos)*DH_ + dc] = acc[t][i];
        }
}

// =====================================================================
// Kernel 5: per-chunk forward + backward of mem_loss -> grads of wq/wk/wv1/wv2
// one workgroup (128 threads, 4 waves) per chunk; 128KB dynamic LDS
// =====================================================================
__global__ void nm_grad_kernel(const float* __restrict__ keys, const float* __restrict__ values,
                               const float* __restrict__ wq,  const float* __restrict__ wk,
                               const float* __restrict__ wv1, const float* __restrict__ wv2,
                               float* __restrict__ gq, float* __restrict__ gk,
                               float* __restrict__ gv1, float* __restrict__ gv2)
{
    extern __shared__ float sm[];
    float* K  = sm;            // x (keys chunk)
    float* T  = sm + 4096;     // target, later temp X
    float* Q  = sm + 8192;
    float* KK = sm + 12288;
    float* V  = sm + 16384;
    float* A  = sm + 20480;    // dots -> attn
    float* H  = sm + 24576;    // hidden, later dattn/ddots
    float* P  = sm + 28672;    // pred -> dpred, later dv

    const int cid = blockIdx.x, tid = threadIdx.x;
    __builtin_prefetch((const void*)wq, 0, 3);
    __builtin_prefetch((const void*)wv2, 0, 3);
    const float* kc = keys   + (size_t)cid*4096;
    const float* vc = values + (size_t)cid*4096;
    for (int e = tid; e < 4096; e += 128) { K[e] = kc[e]; T[e] = vc[e]; }
    __syncthreads();
    // ---- forward ----
    mm64<false,false,false>(Q,  K, wq,  1.f); __syncthreads();
    mm64<false,false,false>(KK, K, wk,  1.f); __syncthreads();
    mm64<false,false,false>(V,  K, wv1, 1.f); __syncthreads();
    mm64<false,true ,false>(A,  Q, KK,  SCALE_); __syncthreads();   // dots
    if (tid < 64) {                                                 // causal softmax
        float* row = A + tid*64;
        float mx = -3.4e38f;
        for (int j = 0; j <= tid; ++j) mx = fmaxf(mx, row[j]);
        float ssum = 0.f;
        for (int j = 0; j <= tid; ++j) { float e2 = __expf(row[j]-mx); row[j]=e2; ssum+=e2; }
        const float inv = 1.f/ssum;
        for (int j = 0; j < 64; ++j) row[j] = (j <= tid) ? row[j]*inv : 0.f;
    }
    __syncthreads();
    mm64<false,false,false>(H, A, V,   1.f); __syncthreads();       // hidden
    mm64<false,false,true >(P, H, wv2, 1.f); __syncthreads();       // pred = silu(H)@wv2
    for (int e = tid; e < 4096; e += 128) P[e] = (P[e] - T[e]) * (2.f/DH_);  // dpred
    __syncthreads();
    // ---- backward ----
    mm64<true ,false,true >(gv2 + (size_t)cid*4096, H, P, 1.f); __syncthreads(); // silu(H)^T@dpred
    mm64<false,true ,false>(T, P, wv2, 1.f); __syncthreads();       // dact
    for (int e = tid; e < 4096; e += 128) {                         // dhidden
        const float hv = H[e], sg = sigmoidf_(hv);
        T[e] *= sg * (1.f + hv * (1.f - sg));
    }
    __syncthreads();
    mm64<true ,false,false>(P, A, T, 1.f); __syncthreads();         // dv = attn^T@dhidden
    mm64<true ,false,false>(gv1 + (size_t)cid*4096, K, P, 1.f); __syncthreads();
    mm64<false,true ,false>(H, T, V, 1.f); __syncthreads();         // dattn
    if (tid < 64) {                                                 // softmax bwd (+fold scale)
        float* da = H + tid*64; float* aa = A + tid*64;
        float sd = 0.f;
        for (int j = 0; j < 64; ++j) sd += da[j]*aa[j];
        for (int j = 0; j < 64; ++j) da[j] = aa[j]*(da[j]-sd)*SCALE_;
    }
    __syncthreads();
    mm64<false,false,false>(T, H, KK, 1.f); __syncthreads();        // dq
    mm64<true ,false,false>(gq + (size_t)cid*4096, K, T, 1.f); __syncthreads();
    mm64<true ,false,false>(T, H, Q, 1.f); __syncthreads();         // dk
    mm64<true ,false,false>(gk + (size_t)cid*4096, K, T, 1.f);
}

// =====================================================================
// Kernel 6: double gated scan over chunks; writes wwu = past + upd in place
// one thread per (weight, bh, element)
// =====================================================================
__global__ void nm_scan_kernel(float* __restrict__ g,
                               const float* __restrict__ lr, const float* __restrict__ mom,
                               const float* __restrict__ dec,
                               const float* __restrict__ wq,  const float* __restrict__ wk,
                               const float* __restrict__ wv1, const float* __restrict__ wv2)
{
    const int idx = blockIdx.x * blockDim.x + threadIdx.x;   // 4*16*4096
    const int e  = idx & 4095;
    const int bh = (idx >> 12) & 15;
    const int w  = idx >> 16;
    const float* base = (w == 0) ? wq : (w == 1) ? wk : (w == 2) ? wv1 : wv2;
    float* gw = g + (size_t)w * NCHUNKS_ * 4096;
    float m = 0.f, u = 0.f;
    for (int ci = 0; ci < NC_; ++ci) {
        const int go = bh*NC_ + ci;
        const size_t off = (size_t)go*4096 + e;
        const float surprise = gw[off] * (-lr[go]);
        m = mom[go]*m + surprise;
        u = (1.f - dec[go])*u + m;
        gw[off] = base[e] + u;       // wwu
    }
}

// =====================================================================
// Kernel 7: retrieval mem_attn per chunk with updated weights
// =====================================================================
__global__ void nm_retr_kernel(const float* __restrict__ qch, const float* __restrict__ wwu,
                               float* __restrict__ vals)
{
    extern __shared__ float sm[];
    float* X  = sm;
    float* Q  = sm + 4096;
    float* KK = sm + 8192;
    float* V  = sm + 12288;
    float* A  = sm + 16384;
    float* H  = sm + 20480;
    float* O  = sm + 24576;
    const int cid = blockIdx.x, tid = threadIdx.x;
    const float* wq  = wwu + (size_t)0*NCHUNKS_*4096 + (size_t)cid*4096;
    const float* wk  = wwu + (size_t)1*NCHUNKS_*4096 + (size_t)cid*4096;
    const float* wv1 = wwu + (size_t)2*NCHUNKS_*4096 + (size_t)cid*4096;
    const float* wv2 = wwu + (size_t)3*NCHUNKS_*4096 + (size_t)cid*4096;
    __builtin_prefetch((const void*)wq, 0, 3);
    __builtin_prefetch((const void*)wv2, 0, 3);
    const float* xc = qch + (size_t)cid*4096;
    for (int e = tid; e < 4096; e += 128) X[e] = xc[e];
    __syncthreads();
    mm64<false,false,false>(Q,  X, wq,  1.f); __syncthreads();
    mm64<false,false,false>(KK, X, wk,  1.f); __syncthreads();
    mm64<false,false,false>(V,  X, wv1, 1.f); __syncthreads();
    mm64<false,true ,false>(A,  Q, KK,  SCALE_); __syncthreads();
    if (tid < 64) {
        float* row = A + tid*64;
        float mx = -3.4e38f;
        for (int j = 0; j <= tid; ++j) mx = fmaxf(mx, row[j]);
        float ssum = 0.f;
        for (int j = 0; j <= tid; ++j) { float e2 = __expf(row[j]-mx); row[j]=e2; ssum+=e2; }
        const float inv = 1.f/ssum;
        for (int j = 0; j < 64; ++j) row[j] = (j <= tid) ? row[j]*inv : 0.f;
    }
    __syncthreads();
    mm64<false,false,false>(H, A, V,   1.f); __syncthreads();
    mm64<false,false,true >(O, H, wv2, 1.f); __syncthreads();
    float* vo = vals + (size_t)cid*4096;
    for (int e = tid; e < 4096; e += 128) vo[e] = O[e];
}

// =====================================================================
// Kernel 8: combined = (vals un-chunkified) @ w_combine
// =====================================================================
__global__ void nm_gemm_comb(const float* __restrict__ vals, const float* __restrict__ w_comb,
                             float* __restrict__ comb)
{
    extern __shared__ float sm[];
    float* At = sm; float* Wt = sm + 4096;
    const int mt = blockIdx.x, nt = blockIdx.y, tid = threadIdx.x;
    v8f acc[4] = {};
    for (int kb = 0; kb < DIM_/64; ++kb) {
        for (int e = tid; e < 4096; e += 128) {
            const int rr = e >> 6, cc = e & 63;
            const int grow = mt*64 + rr;
            const int bb = grow >> 12, j = grow & 4095;
            const int col = kb*64 + cc;
            const int h = col >> 6, dc = col & 63;
            At[e] = vals[(((size_t)(bb*HEADS_ + h)*NC_ + (j >> 6))*CHUNK_ + (j & 63))*DH_ + dc];
            Wt[e] = w_comb[(size_t)(kb*64 + rr)*DIM_ + nt*64 + cc];
        }
        __syncthreads();
        tile_mma(At, Wt, acc);
        __syncthreads();
    }
    const int lane = tid & 31, wave = tid >> 5;
    const int row0 = wave*16 + ((lane >> 4) ? 8 : 0);
    const int l16  = lane & 15;
#pragma unroll
    for (int t = 0; t < 4; ++t)
#pragma unroll
        for (int i = 0; i < 8; ++i) {
            const int grow = mt*64 + row0 + i;
            const int gcol = nt*64 + t*16 + l16;
            comb[(size_t)grow*DIM_ + gcol] = acc[t][i];
        }
}

// =====================================================================
// Kernel 9: post rmsnorm + front-pad-by-CHUNK shift into final output
// =====================================================================
__global__ void nm_post_kernel(const float* __restrict__ comb,
                               const float* __restrict__ w_post,
                               float* __restrict__ out)
{
    const int row = blockIdx.x * blockDim.x + threadIdx.x;   // 0..B*S-1
    if (row >= B_*S_) return;
    const int bb = row >> 12, t = row & 4095;
    float* o = out + (size_t)row * DIM_;
    if (t < CHUNK_) { for (int d = 0; d < DIM_; ++d) o[d] = 0.f; return; }
    const float* x = comb + ((size_t)bb*S_ + (t - CHUNK_)) * DIM_;
    float ss = 0.f;
    for (int d = 0; d < DIM_; ++d) { float v = x[d]; ss += v*v; }
    const float inv = rsqrtf(ss * (1.f/DIM_) + EPS_);
    for (int d = 0; d < DIM_; ++d) o[d] = x[d] * inv * w_post[d];
}

// =====================================================================
// Host launcher
// =====================================================================
extern "C" void kernel_launch(void* const* d_in, const int* in_sizes, int n_in,
                              void* d_out, int out_size, void* d_ws, size_t ws_size,
                              hipStream_t stream)
{
    (void)in_sizes; (void)n_in; (void)out_size; (void)ws_size;
    const float* seq      = (const float*)d_in[0];
    const float* w_store  = (const float*)d_in[1];
    const float* w_retr   = (const float*)d_in[2];
    const float* w_post   = (const float*)d_in[3];
    const float* w_q      = (const float*)d_in[4];
    const float* w_kv     = (const float*)d_in[5];
    const float* w_step   = (const float*)d_in[6];
    const float* w_mom    = (const float*)d_in[7];
    const float* w_decay  = (const float*)d_in[8];
    const float* w_comb   = (const float*)d_in[9];
    const float* wq       = (const float*)d_in[10];
    const float* wk       = (const float*)d_in[11];
    const float* wv1      = (const float*)d_in[12];
    const float* wv2      = (const float*)d_in[13];
    float* out = (float*)d_out;

    // scratch layout (floats); NSF = B*S*DIM = 4,194,304
    const size_t NSF = (size_t)B_ * S_ * DIM_;
    float* ws      = (float*)d_ws;
    float* s       = ws + 0*NSF;        // also reused as vals (chunked) later
    float* r       = ws + 1*NSF;        // also reused as comb later
    float* keys    = ws + 2*NSF;
    float* values  = ws + 3*NSF;
    float* qch     = ws + 4*NSF;
    float* grads   = ws + 5*NSF;        // 4*NSF: gq,gk,gv1,gv2 -> becomes wwu
    float* lr      = ws + 9*NSF;
    float* mom     = lr + NCHUNKS_;
    float* dec     = mom + NCHUNKS_;
    float* valsb   = s;                 // alias (s dead after gemm_kv/gates)
    float* comb    = r;                 // alias (r dead after gemm_q)

    const size_t lds_tile = 2 * 4096 * sizeof(float);     //  32 KB
    const int    lds_grad = 8 * 4096 * sizeof(float);     // 128 KB
    const int    lds_retr = 7 * 4096 * sizeof(float);     // 112 KB
    (void)hipFuncSetAttribute((const void*)nm_grad_kernel,
                              hipFuncAttributeMaxDynamicSharedMemorySize, lds_grad);
    (void)hipFuncSetAttribute((const void*)nm_retr_kernel,
                              hipFuncAttributeMaxDynamicSharedMemorySize, lds_retr);

    nm_norm_kernel<<<dim3((B_*S_)/256), dim3(256), 0, stream>>>(seq, w_store, w_retr, s, r);
    nm_gates_kernel<<<dim3(B_*NC_), dim3(256), 0, stream>>>(s, w_step, w_mom, w_decay, lr, mom, dec);
    nm_gemm_kv<<<dim3((B_*S_)/64, (2*DIM_)/64), dim3(128), lds_tile, stream>>>(s, w_kv, keys, values);
    nm_gemm_q <<<dim3((B_*S_)/64,  DIM_/64),    dim3(128), lds_tile, stream>>>(r, w_q, qch);
    nm_grad_kernel<<<dim3(NCHUNKS_), dim3(128), lds_grad, stream>>>(
        keys, values, wq, wk, wv1, wv2,
        grads + 0*NCHUNKS_*(size_t)4096, grads + 1*NCHUNKS_*(size_t)4096,
        grads + 2*NCHUNKS_*(size_t)4096, grads + 3*NCHUNKS_*(size_t)4096);
    nm_scan_kernel<<<dim3((4*BH_*4096)/256), dim3(256), 0, stream>>>(
        grads, lr, mom, dec, wq, wk, wv1, wv2);
    nm_retr_kernel<<<dim3(NCHUNKS_), dim3(128), lds_retr, stream>>>(qch, grads, valsb);
    nm_gemm_comb<<<dim3((B_*S_)/64, DIM_/64), dim3(128), lds_tile, stream>>>(valsb, w_comb, comb);
    nm_post_kernel<<<dim3((B_*S_)/256), dim3(256), 0, stream>>>(comb, w_post, out);
}